// GATLayer_89567247990859
// MI455X (gfx1250) — compile-verified
//
#include <hip/hip_runtime.h>

#define HEADS 4
#define OUT_F 32
#define IN_F 64
#define EDGE_F 16
#define HF 128           // HEADS*OUT_F
#define NEG_SLOPE 0.2f

typedef __attribute__((ext_vector_type(2))) float v2f;
typedef __attribute__((ext_vector_type(8))) float v8f;

__device__ __forceinline__ v8f wmma_f32_k4(v2f a, v2f b, v8f c) {
  // D = A(16x4 f32) * B(4x16 f32) + C(16x16 f32)
  return __builtin_amdgcn_wmma_f32_16x16x4_f32(
      /*neg_a=*/false, a, /*neg_b=*/false, b,
      /*c_mod=*/(short)0, c, /*reuse_a=*/false, /*reuse_b=*/false);
}

// ---------------------------------------------------------------- K0: zero
__global__ void zero_f32_kernel(float* __restrict__ p, int n) {
  int i = blockIdx.x * blockDim.x + threadIdx.x;
  if (i < n) p[i] = 0.0f;
}

// ------------------------------------------------ K1: ve[h][k] = sum_f W_edge[h*32+f,k]*w_e[f]
__global__ void edge_vec_kernel(const float* __restrict__ W_edge,
                                const float* __restrict__ w_e,
                                float* __restrict__ ve) {
  int t = threadIdx.x;              // 64 threads: h*16 + k
  if (t >= HEADS * EDGE_F) return;
  int h = t >> 4, k = t & 15;
  float s = 0.0f;
  #pragma unroll
  for (int f = 0; f < OUT_F; ++f)
    s += W_edge[(h * OUT_F + f) * EDGE_F + k] * w_e[f];
  ve[t] = s;
}

// ------------------------------------------------ K2: fused node GEMM via WMMA
// One wave per 16-node row tile. Computes [16,64] x [64,256] where the 256
// output columns are 128 of W_lin (-> xp in ws) and 128 of W_res (-> d_out
// preloaded with residual + bias; K5 atomically accumulates messages on top).
__global__ void node_gemm_kernel(const float* __restrict__ x,
                                 const float* __restrict__ Wlin,
                                 const float* __restrict__ Wres,
                                 const float* __restrict__ bias,
                                 float* __restrict__ xp,
                                 float* __restrict__ out,
                                 int n) {
  int wave = (int)((blockIdx.x * blockDim.x + threadIdx.x) >> 5);
  int lane = threadIdx.x & 31;
  int rowBase = wave * 16;
  if (rowBase >= n) return;                 // uniform per-wave exit

  // A-fragment layout (32-bit A 16x4): lanes 0-15 hold row m, K={0,1};
  // lanes 16-31 hold row m, K={2,3}.
  int m = lane & 15;
  int koff = (lane >> 4) * 2;
  const float* xrow = x + (size_t)(rowBase + m) * IN_F + koff;
  v2f afrag[16];
  #pragma unroll
  for (int kk = 0; kk < 16; ++kk)
    afrag[kk] = *(const v2f*)(xrow + kk * 4);

  // 16 column tiles of 16: tiles 0..7 -> W_lin, 8..15 -> W_res.
  for (int ct = 0; ct < 16; ++ct) {
    const float* W = (ct < 8) ? (Wlin + (size_t)(ct * 16) * IN_F)
                              : (Wres + (size_t)((ct - 8) * 16) * IN_F);
    // B-fragment (4x16): lanes 0-15 col n, K={0,1}; lanes 16-31 col n, K={2,3}.
    // out[m,n] = sum_k x[m,k]*W[n,k]  =>  B[k][n] = W[n][k] (row-major W, stride 64)
    const float* wrow = W + (size_t)(lane & 15) * IN_F + koff;
    v8f acc = {};
    #pragma unroll
    for (int kk = 0; kk < 16; ++kk) {
      v2f bfrag = *(const v2f*)(wrow + kk * 4);
      acc = wmma_f32_k4(afrag[kk], bfrag, acc);
    }
    // C/D layout: lanes 0-15 -> col n, rows 0..7 in v[0..7]; lanes 16-31 rows 8..15
    int col = (ct & 7) * 16 + (lane & 15);
    int rowOff = (lane >> 4) * 8;
    if (ct < 8) {
      float* dst = xp + (size_t)rowBase * HF + col;
      #pragma unroll
      for (int r = 0; r < 8; ++r)
        dst[(size_t)(rowOff + r) * HF] = acc[r];
    } else {
      float bv = bias[col];
      float* dst = out + (size_t)rowBase * HF + col;
      #pragma unroll
      for (int r = 0; r < 8; ++r)
        dst[(size_t)(rowOff + r) * HF] = acc[r] + bv;
    }
  }
}

// ------------------------------------------------ K3: a_s/a_t per (node, head)
// One block of 128 threads per node: thread t -> head t/32, feature t%32.
// Wave32 shuffle reduction: each wave reduces one head.
__global__ void attn_scalar_kernel(const float* __restrict__ xp,
                                   const float* __restrict__ w_s,
                                   const float* __restrict__ b_s,
                                   const float* __restrict__ w_t,
                                   const float* __restrict__ b_t,
                                   float* __restrict__ a_s,
                                   float* __restrict__ a_t,
                                   int n) {
  int node = blockIdx.x;
  if (node >= n) return;
  int t = threadIdx.x;              // 0..127
  int h = t >> 5, f = t & 31;
  float v = xp[(size_t)node * HF + t];
  float ps = v * w_s[f];
  float pt = v * w_t[f];
  #pragma unroll
  for (int off = 16; off > 0; off >>= 1) {
    ps += __shfl_down(ps, off, 32);
    pt += __shfl_down(pt, off, 32);
  }
  if (f == 0) {
    a_s[node * HEADS + h] = ps + b_s[0];
    a_t[node * HEADS + h] = pt + b_t[0];
  }
}

// ------------------------------------------------ K4: per-edge exp(logit) + denom
__global__ void edge_logits_kernel(const int* __restrict__ ei,
                                   const float* __restrict__ ea,
                                   const float* __restrict__ a_s,
                                   const float* __restrict__ a_t,
                                   const float* __restrict__ ve,
                                   const float* __restrict__ b_e,
                                   float* __restrict__ expv,
                                   float* __restrict__ denom,
                                   int e_count) {
  int e = blockIdx.x * blockDim.x + threadIdx.x;
  if (e >= e_count) return;
  int src = ei[e];
  int dst = ei[e_count + e];

  float eav[EDGE_F];
  const float4* ea4 = (const float4*)(ea + (size_t)e * EDGE_F);
  #pragma unroll
  for (int q = 0; q < 4; ++q) {
    float4 v4 = ea4[q];
    eav[q * 4 + 0] = v4.x; eav[q * 4 + 1] = v4.y;
    eav[q * 4 + 2] = v4.z; eav[q * 4 + 3] = v4.w;
  }
  float be = b_e[0];
  float ex4[HEADS];
  #pragma unroll
  for (int h = 0; h < HEADS; ++h) {
    float s = be;
    #pragma unroll
    for (int k = 0; k < EDGE_F; ++k) s += eav[k] * ve[h * EDGE_F + k];
    float al = a_s[src * HEADS + h] + a_t[dst * HEADS + h] + s;
    al = (al >= 0.0f) ? al : NEG_SLOPE * al;
    // logits bounded (|al| ~ 1): softmax without max-subtraction is exact to rounding
    float ex = __expf(al);
    ex4[h] = ex;
    atomicAdd(&denom[dst * HEADS + h], ex);
  }
  float4* ev4 = (float4*)(expv + (size_t)e * HEADS);
  *ev4 = make_float4(ex4[0], ex4[1], ex4[2], ex4[3]);
}

// ------------------------------------------------ K5: normalize + weighted scatter
// One wave32 per edge; lane = feature, loop over 4 heads. xp/out are L2-resident
// (25.6 MB each), so gathers and atomics stay in the 192 MB L2.
__global__ void edge_scatter_kernel(const int* __restrict__ ei,
                                    const float* __restrict__ expv,
                                    const float* __restrict__ denom,
                                    const float* __restrict__ xp,
                                    float* __restrict__ out,
                                    int e_count) {
  int e = (int)((blockIdx.x * blockDim.x + threadIdx.x) >> 5);
  int lane = threadIdx.x & 31;
  if (e >= e_count) return;
  int src = ei[e];
  int dst = ei[e_count + e];
  const float* xps = xp + (size_t)src * HF;
  float* outd = out + (size_t)dst * HF;
  #pragma unroll
  for (int h = 0; h < HEADS; ++h) {
    float attn = expv[(size_t)e * HEADS + h] / denom[dst * HEADS + h];
    atomicAdd(&outd[h * OUT_F + lane], attn * xps[h * OUT_F + lane]);
  }
}

// ---------------------------------------------------------------- launcher
extern "C" void kernel_launch(void* const* d_in, const int* in_sizes, int n_in,
                              void* d_out, int out_size, void* d_ws, size_t ws_size,
                              hipStream_t stream) {
  const float* x       = (const float*)d_in[0];
  const int*   ei      = (const int*)  d_in[1];
  const float* ea      = (const float*)d_in[2];
  const float* W_lin   = (const float*)d_in[3];
  const float* w_s     = (const float*)d_in[4];
  const float* b_s     = (const float*)d_in[5];
  const float* w_t     = (const float*)d_in[6];
  const float* b_t     = (const float*)d_in[7];
  const float* W_edge  = (const float*)d_in[8];
  const float* w_e     = (const float*)d_in[9];
  const float* b_e     = (const float*)d_in[10];
  const float* W_res   = (const float*)d_in[11];
  const float* bias    = (const float*)d_in[12];
  float*       out     = (float*)d_out;

  const int N = in_sizes[0] / IN_F;
  const int E = in_sizes[2] / EDGE_F;

  // workspace layout (floats)
  float* ws_f  = (float*)d_ws;
  float* xp    = ws_f;                       // N*128
  float* a_s   = xp    + (size_t)N * HF;     // N*4
  float* a_t   = a_s   + (size_t)N * HEADS;  // N*4
  float* denom = a_t   + (size_t)N * HEADS;  // N*4
  float* ve    = denom + (size_t)N * HEADS;  // 64
  float* expv  = ve    + HEADS * EDGE_F;     // E*4

  // K0: zero softmax denominators
  {
    int n = N * HEADS;
    zero_f32_kernel<<<(n + 255) / 256, 256, 0, stream>>>(denom, n);
  }
  // K1: fold W_edge with w_e
  edge_vec_kernel<<<1, 64, 0, stream>>>(W_edge, w_e, ve);

  // K2: WMMA node GEMM (xp + residual). One wave per 16 rows.
  {
    int waves  = (N + 15) / 16;
    int blocks = (waves + 7) / 8;            // 256 threads = 8 waves
    node_gemm_kernel<<<blocks, 256, 0, stream>>>(x, W_lin, W_res, bias, xp, out, N);
  }
  // K3: per-node per-head attention scalars
  attn_scalar_kernel<<<N, 128, 0, stream>>>(xp, w_s, b_s, w_t, b_t, a_s, a_t, N);

  // K4: edge logits -> expv, denom
  edge_logits_kernel<<<(E + 255) / 256, 256, 0, stream>>>(
      ei, ea, a_s, a_t, ve, b_e, expv, denom, E);

  // K5: normalize + scatter messages into out (one wave per edge)
  {
    int blocks = (E + 7) / 8;                // 8 waves per block
    edge_scatter_kernel<<<blocks, 256, 0, stream>>>(ei, expv, denom, xp, out, E);
  }
}